// PyG_GAT_Paper_26792005992877
// MI455X (gfx1250) — compile-verified
//
#include <hip/hip_runtime.h>

typedef __attribute__((ext_vector_type(2))) float v2f;
typedef __attribute__((ext_vector_type(8))) float v8f;

static constexpr float NEG_SLOPE = 0.2f;

__device__ __forceinline__ unsigned flip_f(float f) {
  unsigned u = __float_as_uint(f);
  return (u >> 31) ? ~u : (u | 0x80000000u);
}
__device__ __forceinline__ float unflip_f(unsigned u) {
  return __uint_as_float((u >> 31) ? (u & 0x7fffffffu) : ~u);
}
__device__ __forceinline__ float lrelu(float x) { return x > 0.f ? x : NEG_SLOPE * x; }
__device__ __forceinline__ float eluf(float x)  { return x > 0.f ? x : expm1f(x); }

// ---------------- zero fill (grid-stride) ----------------
__global__ void zero_kernel(float* __restrict__ p, long long n) {
  long long i = (long long)blockIdx.x * blockDim.x + threadIdx.x;
  long long stride = (long long)gridDim.x * blockDim.x;
  for (; i < n; i += stride) p[i] = 0.f;
}

// ---------------- GEMM1: H1[N,FH] = X[N,FIN] @ W1[FIN,FH], WMMA f32 16x16x4 ----
__global__ __launch_bounds__(256) void gemm1_kernel(
    const float* __restrict__ X, const float* __restrict__ W1,
    float* __restrict__ H1, int N, int K, int FH) {
  int wave = (int)((blockIdx.x * blockDim.x + threadIdx.x) >> 5);
  int lane = threadIdx.x & 31;
  int ntiles = (N + 15) >> 4;
  if (wave >= ntiles) return;
  int half = lane >> 4;          // 0: K=k,k+1   1: K=k+2,k+3
  int mrow = lane & 15;          // A row / B column
  int row = wave * 16 + mrow; if (row >= N) row = N - 1;
  const float* xr = X + (size_t)row * K;
  int wcol = (mrow < FH) ? mrow : 0;
  float wm = (mrow < FH) ? 1.f : 0.f;
  v8f c = {};
  for (int k = 0; k < K; k += 4) {
    int k0 = k + 2 * half;
    v2f a; a.x = xr[k0]; a.y = xr[k0 + 1];
    v2f b; b.x = W1[(size_t)k0 * FH + wcol] * wm;
           b.y = W1[(size_t)(k0 + 1) * FH + wcol] * wm;
    c = __builtin_amdgcn_wmma_f32_16x16x4_f32(false, a, false, b, (short)0, c, false, false);
  }
  if (mrow < FH) {
#pragma unroll
    for (int v = 0; v < 8; v++) {
      int node = wave * 16 + v + 8 * half;   // D layout: VGPR v, lanes 0-15 -> M=v, 16-31 -> M=v+8
      if (node < N) H1[(size_t)node * FH + mrow] = c[v];
    }
  }
}

// ---------------- GEMM2: H2[N,32] = elu(AGG1[N,8]+b1) @ W2[8,32], WMMA ----------
__global__ __launch_bounds__(256) void gemm2_kernel(
    const float* __restrict__ AGG1, const float* __restrict__ b1,
    const float* __restrict__ W2, float* __restrict__ H2, int N, int FH, int FO) {
  int wave = (int)((blockIdx.x * blockDim.x + threadIdx.x) >> 5);
  int lane = threadIdx.x & 31;
  int ntiles = (N + 15) >> 4;
  if (wave >= ntiles) return;
  int half = lane >> 4, mrow = lane & 15;
  int row = wave * 16 + mrow; if (row >= N) row = N - 1;
  v8f c0 = {}, c1 = {};
  for (int k = 0; k < FH; k += 4) {
    int k0 = k + 2 * half;
    v2f a;
    a.x = eluf(AGG1[(size_t)row * FH + k0]     + b1[k0]);
    a.y = eluf(AGG1[(size_t)row * FH + k0 + 1] + b1[k0 + 1]);
    v2f bA, bB;
    bA.x = W2[(size_t)k0 * FO + mrow];            bA.y = W2[(size_t)(k0 + 1) * FO + mrow];
    bB.x = W2[(size_t)k0 * FO + 16 + mrow];       bB.y = W2[(size_t)(k0 + 1) * FO + 16 + mrow];
    c0 = __builtin_amdgcn_wmma_f32_16x16x4_f32(false, a, false, bA, (short)0, c0, false, false);
    c1 = __builtin_amdgcn_wmma_f32_16x16x4_f32(false, a, false, bB, (short)0, c1, false, false);
  }
#pragma unroll
  for (int v = 0; v < 8; v++) {
    int node = wave * 16 + v + 8 * half;
    if (node < N) {
      H2[(size_t)node * FO + mrow]      = c0[v];
      H2[(size_t)node * FO + 16 + mrow] = c1[v];
    }
  }
}

// ---------------- per-node attention projections: Es = H.a_src, Ed = H.a_dst ----
__global__ void attn_proj_kernel(const float* __restrict__ H,
                                 const float* __restrict__ as_, const float* __restrict__ ad_,
                                 float* __restrict__ Es, float* __restrict__ Ed, int N, int F) {
  int i = blockIdx.x * blockDim.x + threadIdx.x;
  if (i >= N) return;
  const float* h = H + (size_t)i * F;
  float s = 0.f, d = 0.f;
  for (int f = 0; f < F; f++) { float v = h[f]; s += v * as_[f]; d += v * ad_[f]; }
  Es[i] = s; Ed[i] = d;
}

// ---------------- edge pass 1: segment max of leaky_relu(Es[src]+Ed[dst]) -------
__global__ void edge_max_kernel(const int* __restrict__ src, const int* __restrict__ dst,
                                long long EC, int N,
                                const float* __restrict__ Es, const float* __restrict__ Ed,
                                unsigned* __restrict__ m) {
  long long e = (long long)blockIdx.x * blockDim.x + threadIdx.x;
  if (e >= EC + N) return;
  int s, d;
  if (e < EC) { s = src[e]; d = dst[e]; } else { s = (int)(e - EC); d = s; }
  float logit = lrelu(Es[s] + Ed[d]);
  atomicMax(&m[d], flip_f(logit));
}

// ---------------- edge pass 2: denom[dst] += exp(logit - m[dst]) ----------------
__global__ void edge_sum_kernel(const int* __restrict__ src, const int* __restrict__ dst,
                                long long EC, int N,
                                const float* __restrict__ Es, const float* __restrict__ Ed,
                                const unsigned* __restrict__ m, float* __restrict__ denom) {
  long long e = (long long)blockIdx.x * blockDim.x + threadIdx.x;
  if (e >= EC + N) return;
  int s, d;
  if (e < EC) { s = src[e]; d = dst[e]; } else { s = (int)(e - EC); d = s; }
  float logit = lrelu(Es[s] + Ed[d]);
  atomicAdd(&denom[d], expf(logit - unflip_f(m[d])));
}

// ---------------- edge pass 3: out[dst] += alpha * H[src]; F lanes per edge -----
template <int F>
__global__ void edge_agg_kernel(const int* __restrict__ src, const int* __restrict__ dst,
                                long long EC, int N,
                                const float* __restrict__ Es, const float* __restrict__ Ed,
                                const unsigned* __restrict__ m, const float* __restrict__ denom,
                                const float* __restrict__ H, float* __restrict__ out) {
  long long tid = (long long)blockIdx.x * blockDim.x + threadIdx.x;
  long long total = (EC + (long long)N) * F;
  if (tid >= total) return;
  long long e = tid / F;
  int f = (int)(tid - e * F);
  int s, d;
  if (e < EC) { s = src[e]; d = dst[e]; } else { s = (int)(e - EC); d = s; }
  float logit = lrelu(Es[s] + Ed[d]);
  float alpha = expf(logit - unflip_f(m[d])) / denom[d];
  atomicAdd(&out[(size_t)d * F + f], alpha * H[(size_t)s * F + f]);
}

// ---------------- final: out += b2 ----------------------------------------------
__global__ void add_bias_kernel(float* __restrict__ out, const float* __restrict__ b,
                                long long total, int F) {
  long long i = (long long)blockIdx.x * blockDim.x + threadIdx.x;
  if (i >= total) return;
  out[i] += b[i % F];
}

static inline unsigned cdiv_u(long long a, long long b) { return (unsigned)((a + b - 1) / b); }

extern "C" void kernel_launch(void* const* d_in, const int* in_sizes, int n_in,
                              void* d_out, int out_size, void* d_ws, size_t ws_size,
                              hipStream_t stream) {
  // inputs: V, E, X, W1, a1_src, a1_dst, b1, W2, a2_src, a2_dst, b2
  const int*   Eidx = (const int*)d_in[1];
  const float* X    = (const float*)d_in[2];
  const float* W1   = (const float*)d_in[3];
  const float* a1s  = (const float*)d_in[4];
  const float* a1d  = (const float*)d_in[5];
  const float* b1   = (const float*)d_in[6];
  const float* W2   = (const float*)d_in[7];
  const float* a2s  = (const float*)d_in[8];
  const float* a2d  = (const float*)d_in[9];
  const float* b2   = (const float*)d_in[10];

  const long long EC = in_sizes[1] / 2;          // 3.2M
  const int FH  = in_sizes[4];                   // 8
  const int FO  = in_sizes[8];                   // 32
  const int FIN = in_sizes[3] / FH;              // 512
  const int N   = in_sizes[2] / FIN;             // 100000
  const int* src = Eidx;
  const int* dst = Eidx + EC;

  // workspace carve-up
  char* w = (char*)d_ws;
  float*    H1     = (float*)w;    w += (size_t)N * FH * sizeof(float);
  float*    Es1    = (float*)w;    w += (size_t)N * sizeof(float);
  float*    Ed1    = (float*)w;    w += (size_t)N * sizeof(float);
  unsigned* m1     = (unsigned*)w; w += (size_t)N * sizeof(unsigned);
  float*    den1   = (float*)w;    w += (size_t)N * sizeof(float);
  float*    agg1   = (float*)w;    w += (size_t)N * FH * sizeof(float);
  float*    H2     = (float*)w;    w += (size_t)N * FO * sizeof(float);
  float*    Es2    = (float*)w;    w += (size_t)N * sizeof(float);
  float*    Ed2    = (float*)w;    w += (size_t)N * sizeof(float);
  unsigned* m2     = (unsigned*)w; w += (size_t)N * sizeof(unsigned);
  float*    den2   = (float*)w;    w += (size_t)N * sizeof(float);
  float*    out    = (float*)d_out;

  const int B = 256;
  const long long TE = EC + N;  // total edges incl. self loops
  const int ntiles = (N + 15) >> 4;

  // init accumulators (m* zero == below flip(-inf); every node has a self-loop)
  zero_kernel<<<1024, B, 0, stream>>>(agg1, (long long)N * FH);
  zero_kernel<<<1024, B, 0, stream>>>(den1, N);
  zero_kernel<<<1024, B, 0, stream>>>((float*)m1, N);
  zero_kernel<<<1024, B, 0, stream>>>(den2, N);
  zero_kernel<<<1024, B, 0, stream>>>((float*)m2, N);
  zero_kernel<<<1024, B, 0, stream>>>(out, (long long)N * FO);

  // layer 1
  gemm1_kernel<<<cdiv_u((long long)ntiles * 32, B), B, 0, stream>>>(X, W1, H1, N, FIN, FH);
  attn_proj_kernel<<<cdiv_u(N, B), B, 0, stream>>>(H1, a1s, a1d, Es1, Ed1, N, FH);
  edge_max_kernel<<<cdiv_u(TE, B), B, 0, stream>>>(src, dst, EC, N, Es1, Ed1, m1);
  edge_sum_kernel<<<cdiv_u(TE, B), B, 0, stream>>>(src, dst, EC, N, Es1, Ed1, m1, den1);
  edge_agg_kernel<8><<<cdiv_u(TE * 8, B), B, 0, stream>>>(src, dst, EC, N, Es1, Ed1, m1, den1, H1, agg1);

  // layer 2 (elu + b1 fused into gemm2's A-matrix load)
  gemm2_kernel<<<cdiv_u((long long)ntiles * 32, B), B, 0, stream>>>(agg1, b1, W2, H2, N, FH, FO);
  attn_proj_kernel<<<cdiv_u(N, B), B, 0, stream>>>(H2, a2s, a2d, Es2, Ed2, N, FO);
  edge_max_kernel<<<cdiv_u(TE, B), B, 0, stream>>>(src, dst, EC, N, Es2, Ed2, m2);
  edge_sum_kernel<<<cdiv_u(TE, B), B, 0, stream>>>(src, dst, EC, N, Es2, Ed2, m2, den2);
  edge_agg_kernel<32><<<cdiv_u(TE * 32, B), B, 0, stream>>>(src, dst, EC, N, Es2, Ed2, m2, den2, H2, out);

  add_bias_kernel<<<cdiv_u((long long)N * FO, B), B, 0, stream>>>(out, b2, (long long)N * FO, FO);
}